// WaterNetCQ2_38268158608159
// MI455X (gfx1250) — compile-verified
//
#include <hip/hip_runtime.h>
#include <math.h>
#include <stdint.h>

// Problem constants (match reference)
#define NT_ 365
#define NS_ 2048
#define NH_ 64
#define NG_ 128
#define NW_ 513      // NH*8 + 1 rows of W
#define WPITCH 528   // NW padded to multiple of 16 (33 column tiles)

typedef float v2f __attribute__((ext_vector_type(2)));
typedef float v8f __attribute__((ext_vector_type(8)));

// GCC-style vector + address-space pointer types matching the builtin's
// actual signature (from the compiler diagnostic):
//   param0: int __attribute__((vector_size(16))) addrspace(1)*  (global src)
//   param1: LDS destination pointer (addrspace(3))
typedef int v4i __attribute__((vector_size(16)));
typedef __attribute__((address_space(1))) v4i* g_v4i_p;
typedef __attribute__((address_space(3))) v4i* l_v4i_p;

// ---------------------------------------------------------------------------
// Kernel 1: w = xc @ W^T + b  via native f32 WMMA (V_WMMA_F32_16X16X4_F32)
// One wave computes one 16x16 tile of w. K = 128 -> 32 WMMA ops per tile.
// Wave-uniform split: tiles 0..31 are fully in-range (no B masking);
// only the last column tile (n0 = 512) masks the padded W rows.
// ---------------------------------------------------------------------------
__global__ __launch_bounds__(128) void wn_gemm_wmma(
    const float* __restrict__ xc,   // [NS, NG]
    const float* __restrict__ Wm,   // [NW, NG]
    const float* __restrict__ bv,   // [NW]
    float* __restrict__ wout)       // [NS, WPITCH]
{
    const int lane = threadIdx.x & 31;
    const int wave = blockIdx.x * (blockDim.x >> 5) + (threadIdx.x >> 5);
    const int NTN  = WPITCH / 16;           // 33 column tiles
    const int mt   = wave / NTN;            // 0..127
    const int nt   = wave - mt * NTN;       // 0..32
    const int m0   = mt << 4;
    const int n0   = nt << 4;
    const int l15  = lane & 15;
    const int hs   = lane >> 4;             // half-wave select

    const int  bcol   = n0 + l15;
    const bool nvalid = (bcol < NW_);
    const float bias  = nvalid ? bv[bcol] : 0.0f;
    v8f acc;
#pragma unroll
    for (int r = 0; r < 8; ++r) acc[r] = bias;

    const float* arow = xc + (m0 + l15) * NG_ + (hs << 1);
    const float* brow = Wm + (nvalid ? bcol : 0) * NG_ + (hs << 1);

    if (n0 + 16 <= NW_) {
        // Full tile: every lane's B row is valid -> pure load + WMMA chain.
#pragma unroll
        for (int k = 0; k < NG_; k += 4) {
            v2f A, B;
            A.x = arow[k];     A.y = arow[k + 1];
            B.x = brow[k];     B.y = brow[k + 1];
            acc = __builtin_amdgcn_wmma_f32_16x16x4_f32(
                false, A, false, B, (short)0, acc, false, false);
        }
    } else {
        // Edge tile (columns 512..527): zero the padded lanes of B.
#pragma unroll
        for (int k = 0; k < NG_; k += 4) {
            v2f A, B;
            A.x = arow[k];     A.y = arow[k + 1];
            float b0 = brow[k], b1 = brow[k + 1];
            B.x = nvalid ? b0 : 0.0f;
            B.y = nvalid ? b1 : 0.0f;
            acc = __builtin_amdgcn_wmma_f32_16x16x4_f32(
                false, A, false, B, (short)0, acc, false, false);
        }
    }

    float* o = wout + (m0 + (hs << 3)) * WPITCH + n0 + l15;
#pragma unroll
    for (int r = 0; r < 8; ++r) o[r * WPITCH] = acc[r];
}

// ---------------------------------------------------------------------------
// Kernel 2: parameter transforms + 365-step register-resident scan.
// One wave per site; each lane owns hidden slots h=lane, h=lane+32.
// The wave's entire forcing slice x[:, site, :] (365 float4 = 5.84 KB) is
// staged into LDS with gfx1250 async-to-LDS loads at kernel start, so the
// latency-bound scan loop reads at LDS latency instead of HBM latency.
// Per-wave ASYNCcnt => no workgroup barrier needed (each wave reads only
// the LDS region it filled itself).
// ---------------------------------------------------------------------------
#define XSLOT 366   // padded 365, float4 slots per wave

#if defined(__has_builtin)
#  if __has_builtin(__builtin_amdgcn_global_load_async_to_lds_b128)
#    define ASYNC_VIA_BUILTIN 1
#  endif
#  if __has_builtin(__builtin_amdgcn_s_wait_asynccnt)
#    define WAITASYNC_VIA_BUILTIN 1
#  endif
#endif

__device__ __forceinline__ float sigf(float v) {
    return 1.0f / (1.0f + __expf(-v));
}
__device__ __forceinline__ float wave_sum(float v) {
#pragma unroll
    for (int off = 16; off > 0; off >>= 1) v += __shfl_xor(v, off, 32);
    return v;
}
__device__ __forceinline__ float wave_max(float v) {
#pragma unroll
    for (int off = 16; off > 0; off >>= 1) v = fmaxf(v, __shfl_xor(v, off, 32));
    return v;
}

__global__ __launch_bounds__(256) void wn_scan(
    const float* __restrict__ x,    // [NT, NS, 4]  (P, E, T1, T2)
    const float* __restrict__ w,    // [NS, WPITCH]
    float* __restrict__ out)        // [NT, NS]
{
    __shared__ float4 sx[8 * XSLOT];            // 46,848 B per workgroup

    const int lane = threadIdx.x & 31;
    const int wv   = threadIdx.x >> 5;          // wave in block: 0..7
    const int site = blockIdx.x * 8 + wv;
    const float*  wr = w + site * WPITCH;
    const float4* xr = (const float4*)x;        // [NT*NS] float4, stride NS per t

    // ---- stage x[:, site] into this wave's LDS region (async DMA) -------
#pragma unroll
    for (int i = 0; i < (NT_ + 31) / 32; ++i) {
        const int t = i * 32 + lane;
        if (t < NT_) {
            const uint64_t ga = (uint64_t)(uintptr_t)(xr + (size_t)t * NS_ + site);
            const uint32_t la = (uint32_t)(uintptr_t)(&sx[wv * XSLOT + t]);
#if defined(ASYNC_VIA_BUILTIN)
            __builtin_amdgcn_global_load_async_to_lds_b128(
                (g_v4i_p)(uintptr_t)ga,   // global source (AS1 v4i*)
                (l_v4i_p)(uintptr_t)la,   // LDS destination (AS3 v4i*)
                0, 0);
#else
            asm volatile("global_load_async_to_lds_b128 %0, %1, off"
                         :: "v"(la), "v"(ga) : "memory");
#endif
        }
    }

    // ---- static per-site parameters (overlaps with async loads) ---------
    const float w512 = wr[8 * NH_];                   // column 512
    const float qb   = fmaxf(w512, 0.0f) * (1.0f / NH_);
    const float vi   = sigf(w512);                    // faithful to source slice

    float gm[2], ge[2], k0[2], k1[2], k2[2], gl[2], kb[2], ga_[2], wk1[2];
#pragma unroll
    for (int i = 0; i < 2; ++i) {
        const int h = lane + 32 * i;
        gm[i]  = __expf(wr[h]) + 1.0f;
        ge[i]  = 2.0f * sigf(wr[1 * NH_ + h]);
        k0[i]  = sigf(wr[3 * NH_ + h]);
        wk1[i] = wr[4 * NH_ + h];
        k1[i]  = sigf(wk1[i]);
        k2[i]  = sigf(wr[5 * NH_ + h]);
        gl[i]  = __expf(wr[6 * NH_ + h]);
        kb[i]  = 0.1f * sigf(wr[7 * NH_ + h]);
    }
    // softmax over the 4nh:5nh slice (same slice as k1), width 64 per site
    const float mx  = wave_max(fmaxf(wk1[0], wk1[1]));
    const float e0  = __expf(wk1[0] - mx);
    const float e1  = __expf(wk1[1] - mx);
    const float inv = 1.0f / wave_sum(e0 + e1);
    ga_[0] = e0 * inv;
    ga_[1] = e1 * inv;

    // ---- wait for this wave's async transfers, then scan from LDS -------
#if defined(WAITASYNC_VIA_BUILTIN)
    __builtin_amdgcn_s_wait_asynccnt(0);
#else
    asm volatile("s_wait_asynccnt 0" ::: "memory");
#endif

    float S0[2] = {0.0f, 0.0f}, H0[2] = {0.0f, 0.0f};
    float H1[2] = {0.0f, 0.0f}, H2[2] = {0.0f, 0.0f};

    const float4* sxe = &sx[wv * XSLOT];
    for (int t = 0; t < NT_; ++t) {
        const float4 xv = sxe[t];               // broadcast ds_load_b128
        const float P = xv.x, E = xv.y, T1 = xv.z, T2 = xv.w;
        const float ta = 0.5f * (T1 + T2);

        float rP;
        if (T2 <= 0.0f)      rP = 0.0f;
        else if (T1 >= 0.0f) rP = 1.0f;
        else rP = 1.0f - acosf((T1 + T2) / (T2 - T1)) * (1.0f / 3.1415f);
        const float ps = (1.0f - rP) * P;   // snow
        const float pl = rP * P;            // liquid

        float y = 0.0f;
#pragma unroll
        for (int i = 0; i < 2; ++i) {
            const float Sm  = fminf(S0[i], fmaxf(ta * gm[i], 0.0f));
            const float G1  = fmaxf(H1[i] + Sm + pl * vi - E * ge[i], 0.0f);
            const float G0  = fmaxf(H0[i] + G1 - gl[i] + pl * (1.0f - vi), 0.0f);
            const float Q0  = G0 * k0[i];
            const float Q1a = fminf(G1, gl[i]) * k1[i];
            const float Q1  = Q1a * (1.0f - kb[i]);
            const float G2  = H2[i] + Q1a * kb[i];
            const float Q2  = G2 * k2[i];
            y += (Q0 + Q1 + Q2 + qb) * ga_[i];
            S0[i] = S0[i] + ps - Sm;
            H0[i] = G0 - Q0;
            H1[i] = fminf(G1 - Q1a, gl[i]);
            H2[i] = G2 - Q2;
        }
        const float ysum = wave_sum(y);
        if (lane == 0) out[t * NS_ + site] = ysum;
    }
}

// ---------------------------------------------------------------------------
extern "C" void kernel_launch(void* const* d_in, const int* in_sizes, int n_in,
                              void* d_out, int out_size, void* d_ws, size_t ws_size,
                              hipStream_t stream) {
    (void)in_sizes; (void)n_in; (void)out_size; (void)ws_size;
    const float* x  = (const float*)d_in[0];   // [365, 2048, 4]
    const float* xc = (const float*)d_in[1];   // [2048, 128]
    const float* Wm = (const float*)d_in[2];   // [513, 128]
    const float* bv = (const float*)d_in[3];   // [513]
    float* wbuf = (float*)d_ws;                // [2048, 528] = 4.3 MB scratch
    float* out  = (float*)d_out;               // [365, 2048]

    // 128 M-tiles * 33 N-tiles = 4224 waves; 4 waves per 128-thread block.
    wn_gemm_wmma<<<1056, 128, 0, stream>>>(xc, Wm, bv, wbuf);
    // One wave per site: 2048 sites / 8 waves per 256-thread block.
    wn_scan<<<NS_ / 8, 256, 0, stream>>>(x, wbuf, out);
}